// Net_15642270892742
// MI455X (gfx1250) — compile-verified
//
#include <hip/hip_runtime.h>
#include <hip/hip_bf16.h>

// Scatter-add: out = A; out[index[e], :] += B[e, :]  (fp32, D=128, wave32)
//
// MI455X reasoning:
//  - ~620 MB total traffic -> ~27us floor at 23.3 TB/s. Pure bandwidth problem.
//  - out (51.2 MB) fits in 192 MB L2 -> atomics resolve in L2. Stream B with
//    non-temporal loads so it does not evict out from L2.
//  - wave32: one wave per B row; lane -> one float4 (global_load_b128,
//    512B/wave coalesced) + 4 non-returning global_atomic_add_f32.

#define DDIM 128

typedef float f4 __attribute__((ext_vector_type(4)));

__global__ __launch_bounds__(256) void scatter_copy_A(const f4* __restrict__ a,
                                                      f4* __restrict__ o,
                                                      int n4) {
  int i = blockIdx.x * 256 + threadIdx.x;
  if (i < n4) {
    // A is read once (streaming): NT load. Store is regular-temporal so the
    // output stays resident in L2 for the atomic phase.
    f4 v = __builtin_nontemporal_load(a + i);
    o[i] = v;
  }
}

__global__ __launch_bounds__(256) void scatter_add_rows(const int* __restrict__ index,
                                                        const float* __restrict__ B,
                                                        float* __restrict__ out,
                                                        int E) {
  unsigned tid = blockIdx.x * 256u + threadIdx.x;
  unsigned e    = tid >> 5;    // one wave32 per element/row
  unsigned lane = tid & 31u;   // lane -> 4 consecutive floats of the row
  if (e >= (unsigned)E) return;   // wave-uniform guard

  // index[e] is wave-uniform: force it into an SGPR so the atomic address
  // math is scalar + lane offset.
  int row = __builtin_amdgcn_readfirstlane(index[e]);

  // Stream B through with a non-temporal 128-bit load (don't pollute L2).
  const f4* bp = (const f4*)(B + (size_t)e * DDIM);
  f4 v = __builtin_nontemporal_load(bp + lane);

  float* dst = out + (size_t)row * DDIM + (size_t)(lane * 4u);

  // Non-returning fp32 atomics -> global_atomic_add_f32 (L2-resident target).
  unsafeAtomicAdd(dst + 0, v.x);
  unsafeAtomicAdd(dst + 1, v.y);
  unsafeAtomicAdd(dst + 2, v.z);
  unsafeAtomicAdd(dst + 3, v.w);
}

extern "C" void kernel_launch(void* const* d_in, const int* in_sizes, int n_in,
                              void* d_out, int out_size, void* d_ws, size_t ws_size,
                              hipStream_t stream) {
  const int*   index = (const int*)d_in[0];    // (E,)   int32
  const float* A     = (const float*)d_in[1];  // (N, D) fp32
  const float* B     = (const float*)d_in[2];  // (E, D) fp32
  float*       out   = (float*)d_out;          // (N, D) fp32

  const int E  = in_sizes[0];
  const int ND = in_sizes[1];  // N * D

  // Phase 1: out = A (vectorized float4 copy).
  const int n4 = ND / 4;
  scatter_copy_A<<<(n4 + 255) / 256, 256, 0, stream>>>((const f4*)A, (f4*)out, n4);

  // Phase 2: scatter-add. One wave (32 lanes) per row of B -> E*32 threads.
  const long long total = (long long)E * 32;
  const int blocks = (int)((total + 255) / 256);
  scatter_add_rows<<<blocks, 256, 0, stream>>>(index, B, out, E);
}